// bilinearDec_66279935312288
// MI455X (gfx1250) — compile-verified
//
#include <hip/hip_runtime.h>

typedef __attribute__((ext_vector_type(2))) float v2f;
typedef __attribute__((ext_vector_type(8))) float v8f;

#define DFEAT 128

// ---------------------------------------------------------------------------
// Kernel 1: u = z @ W using V_WMMA_F32_16X16X4_F32 (full fp32 matrix pipe).
// One workgroup (256 thr = 8 waves) computes 128 rows of u.
// Wave w handles rows [blk*128 + w*16, +16), looping over 8 column tiles.
// A-panel (16 rows x 128 cols of z) is register-resident per wave:
//   per ISA 32-bit A 16x4 layout, lane<16 holds (K=0,1), lane>=16 holds (K=2,3)
//   for row M = lane%16  ->  each lane owns 32 contiguous float2's of its row.
// W lives in LDS (64 KB), read as two ds_load_b32 per K-step (B = 4x16 frag).
// ---------------------------------------------------------------------------
__global__ __launch_bounds__(256) void zw_gemm_wmma(const float* __restrict__ z,
                                                    const float* __restrict__ W,
                                                    float* __restrict__ u,
                                                    int n_nodes) {
    __shared__ float sW[DFEAT * DFEAT];  // 64 KB, row-major W[k][n]

    const int t = threadIdx.x;
    // cooperative, fully-coalesced float4 load of W into LDS
#pragma unroll
    for (int i = 0; i < (DFEAT * DFEAT / 4) / 256; ++i) {
        const int idx4 = i * 256 + t;
        reinterpret_cast<float4*>(sW)[idx4] = reinterpret_cast<const float4*>(W)[idx4];
    }
    __syncthreads();

    const int lane = t & 31;
    const int wave = t >> 5;       // 0..7
    const int m    = lane & 15;    // tile row / tile col index for this lane
    const int hi   = lane >> 4;    // lower/upper half of wave
    const int koff = hi * 2;       // K offset inside each 4-wide K block

    const int r0   = blockIdx.x * 128 + wave * 16;
    int row        = r0 + m;
    // clamp (no branch) so EXEC stays all-ones for WMMA; stores are guarded
    row            = row < n_nodes ? row : n_nodes - 1;
    const float* zrow = z + (size_t)row * DFEAT;

    // register-resident A fragments: 32 K-steps x float2
    v2f a[32];
#pragma unroll
    for (int kk = 0; kk < 32; ++kk) {
        a[kk] = *reinterpret_cast<const v2f*>(zrow + 4 * kk + koff);
    }

    const int mBase = r0 + hi * 8;  // accumulator rows for this lane: mBase + r

#pragma unroll 1
    for (int ct = 0; ct < 8; ++ct) {
        const int n = ct * 16 + m;  // output column for this lane
        v8f acc = {};
#pragma unroll
        for (int kk = 0; kk < 32; ++kk) {
            v2f bf;
            bf.x = sW[(4 * kk + koff) * DFEAT + n];
            bf.y = sW[(4 * kk + koff + 1) * DFEAT + n];
            acc = __builtin_amdgcn_wmma_f32_16x16x4_f32(
                false, a[kk], false, bf, (short)0, acc, false, false);
        }
        // C/D layout: VGPR r -> M = hi*8 + r, N = lane&15 (within tile)
#pragma unroll
        for (int r = 0; r < 8; ++r) {
            const int ro = mBase + r;
            if (ro < n_nodes) u[(size_t)ro * DFEAT + n] = acc[r];
        }
    }
}

// ---------------------------------------------------------------------------
// Kernel 2: y[e] = dot(u[src_e], z[dst_e]) + b.
// 8 lanes per edge, 4 edges per wave. Each 8-lane group covers a full 512 B
// row in 4 x (8 lanes x float4) = 4 contiguous 128 B segments (L2-friendly).
// ---------------------------------------------------------------------------
__global__ __launch_bounds__(256) void edge_score(const float* __restrict__ u,
                                                  const float* __restrict__ z,
                                                  const long long* __restrict__ ei,
                                                  const float* __restrict__ bias,
                                                  float* __restrict__ y,
                                                  int n_edges) {
    const int gt   = blockIdx.x * 256 + threadIdx.x;
    const int wave = gt >> 5;
    const int lane = threadIdx.x & 31;
    const int sub  = lane >> 3;   // edge slot within wave (0..3)
    const int l    = lane & 7;    // lane within edge group
    const int e    = wave * 4 + sub;

    float acc = 0.0f;
    if (e < n_edges) {
        const int src = (int)ei[e];
        const int dst = (int)ei[n_edges + e];
        const float4* ur = reinterpret_cast<const float4*>(u + (size_t)src * DFEAT) + l;
        const float4* zr = reinterpret_cast<const float4*>(z + (size_t)dst * DFEAT) + l;
#pragma unroll
        for (int j = 0; j < 4; ++j) {
            const float4 av = ur[j * 8];
            const float4 bv = zr[j * 8];
            acc += av.x * bv.x + av.y * bv.y + av.z * bv.z + av.w * bv.w;
        }
    }
    // reduce within each aligned 8-lane group (xor masks stay inside group)
    acc += __shfl_xor(acc, 1, 32);
    acc += __shfl_xor(acc, 2, 32);
    acc += __shfl_xor(acc, 4, 32);
    if (l == 0 && e < n_edges) y[e] = acc + bias[0];
}

extern "C" void kernel_launch(void* const* d_in, const int* in_sizes, int n_in,
                              void* d_out, int out_size, void* d_ws, size_t ws_size,
                              hipStream_t stream) {
    // setup_inputs order: z (f32), edge_index (int64), W (f32), b (f32)
    const float*     z  = (const float*)d_in[0];
    const long long* ei = (const long long*)d_in[1];
    const float*     W  = (const float*)d_in[2];
    const float*     b  = (const float*)d_in[3];
    float* y = (float*)d_out;
    float* u = (float*)d_ws;  // 100000*128*4 = 51.2 MB scratch for z @ W

    const int n_nodes = in_sizes[0] / DFEAT;
    const int n_edges = in_sizes[1] / 2;

    const int gemmBlocks = (n_nodes + 127) / 128;
    zw_gemm_wmma<<<gemmBlocks, 256, 0, stream>>>(z, W, u, n_nodes);

    const int edgeBlocks = (n_edges + 31) / 32;  // 32 edges per 256-thr block
    edge_score<<<edgeBlocks, 256, 0, stream>>>(u, z, ei, b, y, n_edges);
}